// StructEXGCNNet_54949811585564
// MI455X (gfx1250) — compile-verified
//
#include <hip/hip_runtime.h>
#include <hip/hip_bf16.h>
#include <stdint.h>

#define DIM   64      // input feature dim
#define ROWF  256     // output row stride in floats: 4 concatenated blocks of 64
#define WPB   8       // wave32s per 256-thread block
#define RING  4       // async-copy pipeline depth per wave (power of 2)

// ---------------------------------------------------------------- utilities
__global__ __launch_bounds__(256) void zero_i32(int* __restrict__ p, int n) {
    int i = blockIdx.x * 256 + threadIdx.x;
    if (i < n) p[i] = 0;
}

__global__ __launch_bounds__(256) void copy_i32(const int* __restrict__ a,
                                                int* __restrict__ b, int n) {
    int i = blockIdx.x * 256 + threadIdx.x;
    if (i < n) b[i] = a[i];
}

// ------------------------------------------------- degree histogram (by dst)
__global__ __launch_bounds__(256) void count_deg(const int* __restrict__ dst,
                                                 int* __restrict__ cnt, int E) {
    int e = blockIdx.x * 256 + threadIdx.x;
    if (e < E) atomicAdd(&cnt[dst[e]], 1);
}

// deg includes self-loop (+1): dinv = 1/sqrt(deg+1)
__global__ __launch_bounds__(256) void compute_dinv(const int* __restrict__ cnt,
                                                    float* __restrict__ dinv, int n) {
    int i = blockIdx.x * 256 + threadIdx.x;
    if (i < n) dinv[i] = rsqrtf((float)cnt[i] + 1.0f);
}

// ------------------------------- single-block exclusive scan (in place) -> rowStart
__global__ __launch_bounds__(1024) void scan_excl(int* __restrict__ data, int n) {
    __shared__ int smem[1024];
    __shared__ int carry;
    const int tid = threadIdx.x;
    if (tid == 0) carry = 0;
    __syncthreads();
    for (int base = 0; base < n; base += 1024) {
        int i = base + tid;
        int v = (i < n) ? data[i] : 0;
        smem[tid] = v;
        __syncthreads();
        for (int off = 1; off < 1024; off <<= 1) {
            int t = (tid >= off) ? smem[tid - off] : 0;
            __syncthreads();
            smem[tid] += t;
            __syncthreads();
        }
        int incl = smem[tid];
        if (i < n) data[i] = (incl - v) + carry;
        __syncthreads();
        if (tid == 1023) carry += incl;
        __syncthreads();
    }
    if (tid == 0) data[n] = carry;   // total == E
}

// ------------------------------------------------------------- CSR fill (by dst)
__global__ __launch_bounds__(256) void fill_csr(const int* __restrict__ src,
                                                const int* __restrict__ dst,
                                                const float* __restrict__ dinv,
                                                int* __restrict__ cursor,
                                                int* __restrict__ colIdx,
                                                float* __restrict__ nrm, int E) {
    int e = blockIdx.x * 256 + threadIdx.x;
    if (e < E) {
        int s = src[e], d = dst[e];
        int pos = atomicAdd(&cursor[d], 1);
        colIdx[pos] = s;
        nrm[pos]    = dinv[s] * dinv[d];
    }
}

// ------------------------------------- x0 block: features -> out[:, 0:64] (float4)
__global__ __launch_bounds__(256) void copy_feat(const float* __restrict__ f,
                                                 float* __restrict__ out, int N) {
    int idx = blockIdx.x * 256 + threadIdx.x;           // N*16 float4 slots
    if (idx >= N * 16) return;
    int node = idx >> 4, q = idx & 15;
    float4 v = reinterpret_cast<const float4*>(f)[(size_t)node * 16 + q];
    reinterpret_cast<float4*>(out + (size_t)node * ROWF)[q] = v;
}

// --------------------------------------------------------------- GCN layer
// One wave32 per node: lane owns dims (2*lane, 2*lane+1). Edge-row gathers are
// pipelined through a RING-deep per-wave LDS ring using the CDNA5 async-copy
// engine (global_load_async_to_lds_b64 / ASYNCcnt). Completion is in-order, so
// with lookahead RING the steady-state wait immediate is exactly RING.
__global__ __launch_bounds__(256) void gcn_layer(const float* __restrict__ xbase,
                                                 float* __restrict__ out,
                                                 const int* __restrict__ rowStart,
                                                 const int* __restrict__ colIdx,
                                                 const float* __restrict__ nrm,
                                                 const float* __restrict__ dinv,
                                                 int N, int inOff, int outOff) {
    __shared__ float buf[WPB * RING * DIM];              // 8 waves x 4 slots x 256B
    const int wave = threadIdx.x >> 5;
    const int lane = threadIdx.x & 31;
    const int node = blockIdx.x * WPB + wave;
    if (node >= N) return;                               // wave-uniform exit

    const float* xin = xbase + inOff;
    const int beg = rowStart[node];
    const int end = rowStart[node + 1];

    float* const slotBase   = &buf[wave * (RING * DIM) + 2 * lane];
    const uint32_t ldsBase  = (uint32_t)(size_t)slotBase;   // LDS byte offset

    float ax = 0.f, ay = 0.f;
    const int cnt = end - beg;
    if (cnt > 0) {
        // ---- prologue: fill the ring (up to RING outstanding) ----
        const int P = (cnt < RING) ? cnt : RING;
        for (int k = 0; k < P; ++k) {
            const float* g = xin + (size_t)colIdx[beg + k] * ROWF + 2 * lane;
            uint32_t ldsa = ldsBase + (uint32_t)(k * (DIM * 4));
            asm volatile("global_load_async_to_lds_b64 %0, %1, off"
                         :: "v"(ldsa), "v"((unsigned long long)(size_t)g) : "memory");
        }
        // ---- steady loop: issue lookahead RING, consume oldest ----
        for (int e = beg; e < end; ++e) {
            float w = nrm[e];
            int nxt = e + RING;
            if (nxt < end) {
                const float* g = xin + (size_t)colIdx[nxt] * ROWF + 2 * lane;
                uint32_t ldsa = ldsBase +
                    (uint32_t)(((nxt - beg) & (RING - 1)) * (DIM * 4));
                asm volatile("global_load_async_to_lds_b64 %0, %1, off"
                             :: "v"(ldsa), "v"((unsigned long long)(size_t)g) : "memory");
                // RING newer loads may stay in flight; edge e is then complete
                asm volatile("s_wait_asynccnt 0x4" ::: "memory");
            } else {
                asm volatile("s_wait_asynccnt 0x0" ::: "memory");   // drain tail
            }
            const float* s = slotBase + ((e - beg) & (RING - 1)) * DIM;
            float2 v = *reinterpret_cast<const float2*>(s);          // ds_load_b64
            ax = fmaf(w, v.x, ax);
            ay = fmaf(w, v.y, ay);
        }
    }
    // self-loop term + ReLU
    float di = dinv[node];
    float coef = di * di;
    float2 xv = *reinterpret_cast<const float2*>(xin + (size_t)node * ROWF + 2 * lane);
    ax = fmaxf(fmaf(coef, xv.x, ax), 0.f);
    ay = fmaxf(fmaf(coef, xv.y, ay), 0.f);
    float2 r; r.x = ax; r.y = ay;
    *reinterpret_cast<float2*>(out + (size_t)node * ROWF + outOff + 2 * lane) = r;
}

// ------------------------------------------------------------------ launcher
extern "C" void kernel_launch(void* const* d_in, const int* in_sizes, int n_in,
                              void* d_out, int out_size, void* d_ws, size_t ws_size,
                              hipStream_t stream) {
    (void)n_in; (void)out_size; (void)ws_size;
    const float* features = (const float*)d_in[0];       // [N, 64] f32
    const int*   edge     = (const int*)d_in[1];         // [2, E]  int32
    float*       out      = (float*)d_out;               // [N, 256] f32

    const int N = in_sizes[0] / DIM;
    const int E = in_sizes[1] / 2;
    const int* src = edge;
    const int* dst = edge + E;

    auto al = [](size_t x) { return (x + 255) & ~(size_t)255; };
    char* w = (char*)d_ws;
    size_t off = 0;
    int*   rowStart = (int*)(w + off);   off += al((size_t)(N + 1) * 4);
    int*   cursor   = (int*)(w + off);   off += al((size_t)N * 4);
    float* dinv     = (float*)(w + off); off += al((size_t)N * 4);
    int*   colIdx   = (int*)(w + off);   off += al((size_t)E * 4);
    float* nrm      = (float*)(w + off); off += al((size_t)E * 4);

    const int gN1 = (N + 1 + 255) / 256;
    const int gN  = (N + 255) / 256;
    const int gE  = (E + 255) / 256;

    // Build normalization + CSR-by-dst (reused by all 3 layers)
    zero_i32    <<<gN1, 256, 0, stream>>>(rowStart, N + 1);
    count_deg   <<<gE,  256, 0, stream>>>(dst, rowStart, E);
    compute_dinv<<<gN,  256, 0, stream>>>(rowStart, dinv, N);
    scan_excl   <<<1, 1024, 0, stream>>>(rowStart, N);
    copy_i32    <<<gN,  256, 0, stream>>>(rowStart, cursor, N);
    fill_csr    <<<gE,  256, 0, stream>>>(src, dst, dinv, cursor, colIdx, nrm, E);

    // x0 block
    copy_feat   <<<(N * 16 + 255) / 256, 256, 0, stream>>>(features, out, N);

    // 3 propagation layers chained through d_out column blocks
    const int gL = (N + WPB - 1) / WPB;
    for (int l = 0; l < 3; ++l) {
        gcn_layer<<<gL, 256, 0, stream>>>(out, out, rowStart, colIdx, nrm, dinv,
                                          N, l * DIM, (l + 1) * DIM);
    }
}